// CRFClassifier_66331474919685
// MI455X (gfx1250) — compile-verified
//
#include <hip/hip_runtime.h>
#include <hip/hip_bf16.h>

typedef float v2f __attribute__((ext_vector_type(2)));
typedef float v8f __attribute__((ext_vector_type(8)));

#define B_ 128
#define T_ 512
#define H_ 1024
#define L_ 23
#define EM_STRIDE 32
#define NEG_BIG (-1.0e30f)

// ---------------------------------------------------------------------------
// Kernel 1: emissions = HS(65536x1024) @ W(1024x23) + b, N padded to 32.
// 512 blocks x 256 threads (8 waves). Each wave: 16 rows x 32 cols.
// f32 WMMA (16x16x4): exact fp32, workload is HBM-bound so no precision trade.
// ---------------------------------------------------------------------------
__global__ void __launch_bounds__(256)
emissions_wmma_kernel(const float* __restrict__ hs, const float* __restrict__ W,
                      const float* __restrict__ bias, float* __restrict__ em) {
  extern __shared__ float sW[];  // 512 groups * 65 floats (bank-conflict pad)
  const int tid = threadIdx.x;
  // Stage W into LDS: f(k,n) = (k>>1)*65 + 2n + (k&1)
  // -> lane's B-frag (W[k][n], W[k+1][n]) is one contiguous b64.
  for (int idx = tid; idx < H_ * 32; idx += 256) {
    int k = idx >> 5, n = idx & 31;
    float v = (n < L_) ? W[k * L_ + n] : 0.0f;
    sW[(k >> 1) * 65 + 2 * n + (k & 1)] = v;
  }
  __syncthreads();

  const int wid = tid >> 5, lane = tid & 31;
  const int rowBase = blockIdx.x * 128 + wid * 16;
  const int hi = lane >> 4;          // K-half select (A layout: lanes 16-31 = K+2)
  const int n0 = lane & 15;
  const float* aptr = hs + (size_t)(rowBase + n0) * H_ + 2 * hi;

  v8f c0 = {};
  v8f c1 = {};
#pragma unroll 4
  for (int kk = 0; kk < H_ / 4; ++kk) {
    v2f a  = *(const v2f*)(aptr + kk * 4);                    // global_load_b64
    const float* bp = sW + (kk * 2 + hi) * 65 + 2 * n0;
    v2f b0 = *(const v2f*)(bp);                               // ds_load_b64 (even banks / odd banks)
    v2f b1 = *(const v2f*)(bp + 32);
    c0 = __builtin_amdgcn_wmma_f32_16x16x4_f32(false, a, false, b0, (short)0, c0, false, false);
    c1 = __builtin_amdgcn_wmma_f32_16x16x4_f32(false, a, false, b1, (short)0, c1, false, false);
  }

  float bias0 = bias[n0];                                     // n0 <= 15 < 23
  float bias1 = (n0 + 16 < L_) ? bias[n0 + 16] : 0.0f;
  // C/D layout: vgpr i -> row M = i + 8*hi, col N = lane&15
#pragma unroll
  for (int i = 0; i < 8; ++i) {
    size_t m = (size_t)(rowBase + hi * 8 + i);
    em[m * EM_STRIDE + n0]      = c0[i] + bias0;
    em[m * EM_STRIDE + n0 + 16] = c1[i] + bias1;
  }
}

// ---------------------------------------------------------------------------
// Reduction helpers (wave32)
// ---------------------------------------------------------------------------
__device__ __forceinline__ float wave_sum(float v) {
#pragma unroll
  for (int s = 16; s >= 1; s >>= 1) v += __shfl_xor(v, s, 32);
  return v;
}
__device__ __forceinline__ float wave_max(float v) {
#pragma unroll
  for (int s = 16; s >= 1; s >>= 1) v = fmaxf(v, __shfl_xor(v, s, 32));
  return v;
}
__device__ __forceinline__ int wave_isum(int v) {
#pragma unroll
  for (int s = 16; s >= 1; s >>= 1) v += __shfl_xor(v, s, 32);
  return v;
}

template <int N>
__device__ __forceinline__ float tree_max(float (&v)[N]) {
#pragma unroll
  for (int s = 1; s < N; s *= 2)
#pragma unroll
    for (int i = 0; i + s < N; i += 2 * s) v[i] = fmaxf(v[i], v[i + s]);
  return v[0];
}
template <int N>
__device__ __forceinline__ float tree_sum(float (&v)[N]) {
#pragma unroll
  for (int s = 1; s < N; s *= 2)
#pragma unroll
    for (int i = 0; i + s < N; i += 2 * s) v[i] += v[i + s];
  return v[0];
}

// ---------------------------------------------------------------------------
// Kernel 2: CRF. One block (64 thr = 2 waves) per sequence.
// wave 0: forward algorithm (log Z) + gold-path score  -> llh[b]
// wave 1: Viterbi recursion + LDS backpointers + backtrace -> pred tags
// ---------------------------------------------------------------------------
__global__ void __launch_bounds__(64)
crf_kernel(const float* __restrict__ em, const int* __restrict__ amask,
           const long long* __restrict__ labels,
           const float* __restrict__ stT, const float* __restrict__ enT,
           const float* __restrict__ trans,
           float* __restrict__ llh, float* __restrict__ pred) {
  __shared__ unsigned char hist[(T_ - 1) * 32];  // backpointers (Viterbi)

  const int b = blockIdx.x;
  const int wid = threadIdx.x >> 5;
  const int lane = threadIdx.x & 31;
  const bool act = lane < L_;
  const float* emb = em + (size_t)b * T_ * EM_STRIDE;
  const int* mk = amask + b * T_;

  // Each lane l' caches the incoming-transition column trans[:, l'].
  float tr[L_];
#pragma unroll
  for (int l = 0; l < L_; ++l) tr[l] = act ? trans[l * L_ + lane] : 0.0f;
  const float st = act ? stT[lane] : NEG_BIG;
  const float en = act ? enT[lane] : NEG_BIG;

  if (wid == 0) {
    // ---------------- forward recursion ----------------
    float alpha = act ? (st + emb[lane]) : NEG_BIG;
    for (int t = 1; t < T_; ++t) {
      const float e = act ? emb[t * EM_STRIDE + lane] : 0.0f;
      const int m = mk[t];
      float vv[L_], tm[L_];
#pragma unroll
      for (int l = 0; l < L_; ++l) { vv[l] = __shfl(alpha, l, 32) + tr[l]; tm[l] = vv[l]; }
      const float mx = tree_max(tm);
      float ex[L_];
#pragma unroll
      for (int l = 0; l < L_; ++l) ex[l] = __expf(vv[l] - mx);
      const float nxt = mx + __logf(tree_sum(ex)) + e;
      alpha = (m && act) ? nxt : alpha;
    }
    // log Z = logsumexp(alpha + end)
    const float x = act ? (alpha + en) : NEG_BIG;
    const float M = wave_max(x);
    const float S = wave_sum(act ? __expf(x - M) : 0.0f);
    const float log_z = M + __logf(S);

    // ---------------- gold-path score (parallel over t) ----------------
    float part = 0.0f;
    int cnt = 0;
    for (int t = lane; t < T_; t += 32) {
      const int lab = (int)labels[b * T_ + t];
      const float e = emb[(size_t)t * EM_STRIDE + lab];
      if (t == 0) {
        part += stT[lab] + e;
      } else if (mk[t]) {
        const int pl = (int)labels[b * T_ + t - 1];
        part += trans[pl * L_ + lab] + e;
      }
      cnt += (mk[t] != 0);
    }
    part = wave_sum(part);
    cnt = wave_isum(cnt);
    const int last_tag = (int)labels[b * T_ + (cnt - 1)];
    part += enT[last_tag];
    if (lane == 0) llh[b] = part - log_z;
  } else {
    // ---------------- Viterbi recursion ----------------
    float sc = act ? (st + emb[lane]) : NEG_BIG;
    for (int t = 1; t < T_; ++t) {
      const float e = act ? emb[t * EM_STRIDE + lane] : 0.0f;
      const int m = mk[t];
      float bv[L_];
      int bi[L_];
#pragma unroll
      for (int l = 0; l < L_; ++l) { bv[l] = __shfl(sc, l, 32) + tr[l]; bi[l] = l; }
      // argmax tree, first-max tie-break (matches jnp.argmax)
#pragma unroll
      for (int s = 1; s < L_; s *= 2)
#pragma unroll
        for (int i = 0; i + s < L_; i += 2 * s)
          if (bv[i + s] > bv[i]) { bv[i] = bv[i + s]; bi[i] = bi[i + s]; }
      hist[(t - 1) * 32 + lane] = (unsigned char)(m ? bi[0] : lane);
      const float ns = bv[0] + e;
      sc = (m && act) ? ns : sc;
    }
    // terminal argmax across lanes (lowest index wins ties)
    float y = act ? (sc + en) : NEG_BIG;
    int yi = lane;
#pragma unroll
    for (int s = 16; s >= 1; s >>= 1) {
      const float ov = __shfl_xor(y, s, 32);
      const int oi = __shfl_xor(yi, s, 32);
      if (ov > y || (ov == y && oi < yi)) { y = ov; yi = oi; }
    }
    // backtrace from LDS (DS ops are in-order within a wave)
    if (lane == 0) {
      int tag = yi;
      pred[(size_t)b * T_ + (T_ - 1)] = mk[T_ - 1] ? (float)tag : 0.0f;
      for (int h = T_ - 2; h >= 0; --h) {
        tag = hist[h * 32 + tag];
        pred[(size_t)b * T_ + h] = mk[h] ? (float)tag : 0.0f;
      }
    }
  }
}

// ---------------------------------------------------------------------------
// Kernel 3: loss = -mean(llh)
// ---------------------------------------------------------------------------
__global__ void __launch_bounds__(32)
loss_kernel(const float* __restrict__ llh, float* __restrict__ out) {
  const int lane = threadIdx.x;
  float s = 0.0f;
  for (int i = lane; i < B_; i += 32) s += llh[i];
  s = wave_sum(s);
  if (lane == 0) out[0] = -s / (float)B_;
}

extern "C" void kernel_launch(void* const* d_in, const int* in_sizes, int n_in,
                              void* d_out, int out_size, void* d_ws, size_t ws_size,
                              hipStream_t stream) {
  const float* hs        = (const float*)d_in[0];      // (128,512,1024) f32
  const int* amask       = (const int*)d_in[1];        // (128,512) i32
  const long long* labels = (const long long*)d_in[2]; // (128,512) i64
  const float* W         = (const float*)d_in[3];      // (1024,23) f32
  const float* bias      = (const float*)d_in[4];      // (23,) f32
  const float* stT       = (const float*)d_in[5];      // (23,) f32
  const float* enT       = (const float*)d_in[6];      // (23,) f32
  const float* trans     = (const float*)d_in[7];      // (23,23) f32

  float* em  = (float*)d_ws;                           // 65536*32 f32 = 8 MB
  float* llh = em + (size_t)B_ * T_ * EM_STRIDE;       // 128 f32
  float* out = (float*)d_out;                          // [loss, pred(65536)]

  emissions_wmma_kernel<<<512, 256, 512 * 65 * sizeof(float), stream>>>(hs, W, bias, em);
  crf_kernel<<<B_, 64, 0, stream>>>(em, amask, labels, stT, enT, trans, llh, out + 1);
  loss_kernel<<<1, 32, 0, stream>>>(llh, out);
}